// GATEncoder_22110491639905
// MI455X (gfx1250) — compile-verified
//
#include <hip/hip_runtime.h>

#define N_NODES 50000
#define N_EDGES 800000
#define ETOT    850000          // edges + self loops
#define DMODEL  128
#define EPS     1e-5f
#define SLOPE   0.2f

typedef __attribute__((ext_vector_type(2))) float v2f;
typedef __attribute__((ext_vector_type(8))) float v8f;

// ---------------------------------------------------------------------------
// z = X @ W  (f32, full precision via V_WMMA_F32_16X16X4_F32)
// Block = 256 threads = 8 waves; wave w owns 16-col tile w (8*16 = 128 cols).
// Grid.x = N/16 row tiles (50000 = 3125*16 exactly, no tail).
// Fragment layouts per CDNA5 ISA 7.12.2:
//   A (16x4 MxK): VGPR j, lane L -> (M = L&15, K = j + 2*(L>>4))
//   B (4x16 KxN): VGPR j, lane L -> (K = j + 2*(L>>4), N = L&15)
//   C/D (16x16):  VGPR i, lane L -> (M = i + 8*(L>>4), N = L&15)
// ---------------------------------------------------------------------------
__global__ __launch_bounds__(256) void gat_gemm_wmma_f32(
    const float* __restrict__ X, const float* __restrict__ W,
    float* __restrict__ Z) {
  const int lane = threadIdx.x & 31;
  const int wave = threadIdx.x >> 5;            // 0..7 -> column tile
  const int half = lane >> 4;                   // 0/1
  const int l15  = lane & 15;
  const int rowBase = blockIdx.x * 16;
  const int nCol = wave * 16 + l15;

  const float* Arow = X + (size_t)(rowBase + l15) * DMODEL;

  v8f acc = {};
  #pragma unroll 4
  for (int k0 = 0; k0 < DMODEL; k0 += 4) {
    const int ka = k0 + 2 * half;
    v2f a, b;
    a.x = Arow[ka];
    a.y = Arow[ka + 1];
    b.x = W[(size_t)ka * DMODEL + nCol];
    b.y = W[(size_t)(ka + 1) * DMODEL + nCol];
    acc = __builtin_amdgcn_wmma_f32_16x16x4_f32(
        /*neg_a=*/false, a, /*neg_b=*/false, b,
        /*c_mod=*/(short)0, acc, /*reuse_a=*/false, /*reuse_b=*/false);
  }
  #pragma unroll
  for (int i = 0; i < 8; ++i) {
    const int r = rowBase + i + 8 * half;
    Z[(size_t)r * DMODEL + nCol] = acc[i];
  }
}

// el[n,h] = sum_d z[n,h*DH+d]*al[h,d] ; er likewise
__global__ void gat_attn_proj(const float* __restrict__ Z,
                              const float* __restrict__ al,
                              const float* __restrict__ ar,
                              float* __restrict__ el, float* __restrict__ er,
                              int H, int DH) {
  const int i = blockIdx.x * blockDim.x + threadIdx.x;   // over N*H
  if (i >= N_NODES * H) return;
  const int n = i / H, h = i % H;
  const float* z = Z + (size_t)n * DMODEL + h * DH;
  const float* a = al + h * DH;
  const float* b = ar + h * DH;
  float sl = 0.f, sr = 0.f;
  for (int d = 0; d < DH; ++d) { const float v = z[d]; sl += v * a[d]; sr += v * b[d]; }
  el[i] = sl;
  er[i] = sr;
}

__global__ void gat_fill_f32(float* __restrict__ p, float v, int n) {
  const int i = blockIdx.x * blockDim.x + threadIdx.x;
  if (i < n) p[i] = v;
}

// float atomic max via monotonic int mapping (init must be -inf)
__device__ __forceinline__ void atomicMaxF(float* addr, float v) {
  if (v >= 0.f) atomicMax((int*)addr, __float_as_int(v));
  else          atomicMin((unsigned int*)addr, (unsigned int)__float_as_int(v));
}

// pass 1: e = leaky_relu(el[src]+er[dst]); store e; m[dst] = max(m[dst], e)
__global__ void gat_edge_max(const int* __restrict__ src, const int* __restrict__ dst,
                             const float* __restrict__ el, const float* __restrict__ er,
                             float* __restrict__ eedge, float* __restrict__ m, int H) {
  const int i = blockIdx.x * blockDim.x + threadIdx.x;
  if (i >= ETOT) return;
  const int s = (i < N_EDGES) ? src[i] : (i - N_EDGES);
  const int d = (i < N_EDGES) ? dst[i] : (i - N_EDGES);
  for (int h = 0; h < H; ++h) {
    const float x = el[s * H + h] + er[d * H + h];
    const float e = (x > 0.f) ? x : SLOPE * x;
    eedge[(size_t)i * H + h] = e;
    atomicMaxF(&m[d * H + h], e);
  }
}

// pass 2: ex = exp(e - m[dst]); overwrite eedge with ex; ssum[dst] += ex
__global__ void gat_edge_expsum(const int* __restrict__ dst,
                                float* __restrict__ eedge,
                                const float* __restrict__ m,
                                float* __restrict__ ssum, int H) {
  const int i = blockIdx.x * blockDim.x + threadIdx.x;
  if (i >= ETOT) return;
  const int d = (i < N_EDGES) ? dst[i] : (i - N_EDGES);
  for (int h = 0; h < H; ++h) {
    const float ex = __expf(eedge[(size_t)i * H + h] - m[d * H + h]);
    eedge[(size_t)i * H + h] = ex;
    atomicAdd(&ssum[d * H + h], ex);
  }
}

// pass 3: wave per edge; lane covers 4 contiguous channels of the 128-wide row
__global__ __launch_bounds__(256) void gat_edge_aggregate(
    const int* __restrict__ src, const int* __restrict__ dst,
    const float* __restrict__ eedge, const float* __restrict__ ssum,
    const float* __restrict__ Z, float* __restrict__ agg, int H, int DH) {
  const int e = (int)((blockIdx.x * (size_t)blockDim.x + threadIdx.x) >> 5);
  const int lane = threadIdx.x & 31;
  if (e >= ETOT) return;
  const int s = (e < N_EDGES) ? src[e] : (e - N_EDGES);
  const int d = (e < N_EDGES) ? dst[e] : (e - N_EDGES);
  const int c = lane * 4;                 // channel base (0..124)
  const int h = c / DH;                   // head for these channels
  const float alpha = eedge[(size_t)e * H + h] / ssum[d * H + h];
  const float4 zv = *(const float4*)(Z + (size_t)s * DMODEL + c);
  float* outp = agg + (size_t)d * DMODEL + c;
  atomicAdd(outp + 0, alpha * zv.x);
  atomicAdd(outp + 1, alpha * zv.y);
  atomicAdd(outp + 2, alpha * zv.z);
  atomicAdd(outp + 3, alpha * zv.w);
}

// per-channel sums of (x + bias) for batch-norm (biased stats)
__global__ __launch_bounds__(128) void gat_bn_stats(
    const float* __restrict__ X, const float* __restrict__ bias,
    float* __restrict__ sum, float* __restrict__ sumsq) {
  const int c = threadIdx.x;              // 0..127
  const float b = bias[c];
  float s = 0.f, s2 = 0.f;
  for (int n = blockIdx.x; n < N_NODES; n += gridDim.x) {
    const float v = X[(size_t)n * DMODEL + c] + b;
    s += v; s2 += v * v;
  }
  atomicAdd(&sum[c], s);
  atomicAdd(&sumsq[c], s2);
}

// out = relu(((x + bias) - mu) * rsqrt(var+eps) * g + be)   (in-place OK)
__global__ void gat_bn_apply(const float* __restrict__ X,
                             const float* __restrict__ bias,
                             const float* __restrict__ sum,
                             const float* __restrict__ sumsq,
                             const float* __restrict__ g,
                             const float* __restrict__ be,
                             float* __restrict__ out) {
  const int i = blockIdx.x * blockDim.x + threadIdx.x;
  if (i >= N_NODES * DMODEL) return;
  const int c = i & (DMODEL - 1);
  const float mu = sum[c] * (1.0f / N_NODES);
  const float var = sumsq[c] * (1.0f / N_NODES) - mu * mu;
  const float v = (X[i] + bias[c] - mu) * rsqrtf(var + EPS) * g[c] + be[c];
  out[i] = v > 0.f ? v : 0.f;
}

extern "C" void kernel_launch(void* const* d_in, const int* in_sizes, int n_in,
                              void* d_out, int out_size, void* d_ws, size_t ws_size,
                              hipStream_t stream) {
  (void)in_sizes; (void)n_in; (void)out_size; (void)ws_size;

  const float* feat = (const float*)d_in[0];
  const int*   src  = (const int*)d_in[1];
  const int*   dst  = (const int*)d_in[2];

  // workspace carve-out (floats)
  float* ws    = (float*)d_ws;
  float* z     = ws;                                   // N*128
  float* agg   = z    + (size_t)N_NODES * DMODEL;      // N*128
  float* el    = agg  + (size_t)N_NODES * DMODEL;      // N*4 (max H)
  float* er    = el   + (size_t)N_NODES * 4;
  float* mmax  = er   + (size_t)N_NODES * 4;
  float* ssum  = mmax + (size_t)N_NODES * 4;
  float* eedge = ssum + (size_t)N_NODES * 4;           // ETOT*4 (max H)
  float* stats = eedge + (size_t)ETOT * 4;             // 256 (sum | sumsq)

  const int T = 256;
  const int ND = N_NODES * DMODEL;

  auto layer = [&](const float* x_in, const float* W, const float* al,
                   const float* ar, const float* b, const float* g,
                   const float* be, int H, int DH, float* out) {
    // 1) z = x_in @ W  (WMMA f32)
    gat_gemm_wmma_f32<<<N_NODES / 16, 256, 0, stream>>>(x_in, W, z);
    // 2) attention logits per node
    const int nh = N_NODES * H;
    gat_attn_proj<<<(nh + T - 1) / T, T, 0, stream>>>(z, al, ar, el, er, H, DH);
    // 3) init segment max / sum
    gat_fill_f32<<<(nh + T - 1) / T, T, 0, stream>>>(mmax, -__builtin_inff(), nh);
    gat_fill_f32<<<(nh + T - 1) / T, T, 0, stream>>>(ssum, 0.f, nh);
    // 4) edge softmax
    gat_edge_max<<<(ETOT + T - 1) / T, T, 0, stream>>>(src, dst, el, er, eedge, mmax, H);
    gat_edge_expsum<<<(ETOT + T - 1) / T, T, 0, stream>>>(dst, eedge, mmax, ssum, H);
    // 5) aggregate messages (wave per edge)
    gat_fill_f32<<<(ND + T - 1) / T, T, 0, stream>>>(agg, 0.f, ND);
    gat_edge_aggregate<<<(int)(((size_t)ETOT * 32 + T - 1) / T), T, 0, stream>>>(
        src, dst, eedge, ssum, z, agg, H, DH);
    // 6) bias + batchnorm(+relu)
    gat_fill_f32<<<1, 256, 0, stream>>>(stats, 0.f, 256);
    gat_bn_stats<<<256, 128, 0, stream>>>(agg, b, stats, stats + DMODEL);
    gat_bn_apply<<<(ND + T - 1) / T, T, 0, stream>>>(agg, b, stats, stats + DMODEL,
                                                     g, be, out);
  };

  // inputs: feat,src,dst, then per-layer {W, al, ar, b, g, be}
  const float* W1  = (const float*)d_in[3];
  const float* al1 = (const float*)d_in[4];
  const float* ar1 = (const float*)d_in[5];
  const float* b1  = (const float*)d_in[6];
  const float* g1  = (const float*)d_in[7];
  const float* be1 = (const float*)d_in[8];
  const float* W2  = (const float*)d_in[9];
  const float* al2 = (const float*)d_in[10];
  const float* ar2 = (const float*)d_in[11];
  const float* b2  = (const float*)d_in[12];
  const float* g2  = (const float*)d_in[13];
  const float* be2 = (const float*)d_in[14];
  const float* W3  = (const float*)d_in[15];
  const float* al3 = (const float*)d_in[16];
  const float* ar3 = (const float*)d_in[17];
  const float* b3  = (const float*)d_in[18];
  const float* g3  = (const float*)d_in[19];
  const float* be3 = (const float*)d_in[20];

  // layer 1: feat -> agg (BN in place; agg becomes next input)
  layer(feat, W1, al1, ar1, b1, g1, be1, 4, 32, agg);
  // layer 2: agg -> agg
  layer(agg, W2, al2, ar2, b2, g2, be2, 4, 32, agg);
  // layer 3: agg -> d_out (single head, 128 wide)
  layer(agg, W3, al3, ar3, b3, g3, be3, 1, 128, (float*)d_out);
}